// PositionMHAttention_12103217840330
// MI455X (gfx1250) — compile-verified
//
#include <hip/hip_runtime.h>
#include <math.h>

// ---------------------------------------------------------------------------
// PositionMHAttention for MI455X (gfx1250, wave32, WMMA).
// Strategy: the 128 MiB attention matrix fits in the 192 MB L2, so run a
// multi-pass pipeline through global memory (L2-resident) with fp32 WMMA
// (V_WMMA_F32_16X16X4_F32) for all three GEMM stages. The P@V stage stages
// V chunks into LDS via the CDNA5 async global->LDS path (ASYNCcnt) when the
// toolchain exposes the builtins, with a plain load/ds-store fallback.
// ---------------------------------------------------------------------------

typedef __attribute__((ext_vector_type(2))) float v2f;
typedef __attribute__((ext_vector_type(4))) float v4f;
typedef __attribute__((ext_vector_type(8))) float v8f;
typedef __attribute__((ext_vector_type(4))) int v4i;

#define GLOBAL_AS __attribute__((address_space(1)))
#define LDS_AS __attribute__((address_space(3)))

#define NT 1024            // tokens (N == M)
#define NB 4               // batch
#define NC 256             // channels
#define NH 8               // heads
#define DH 32              // head dim
#define ROWS (NT * NB)     // 4096 projection rows
#define XSIZE ((size_t)NT * NB * NC)  // x output elems; attn follows in d_out

#define VCH 256            // V rows staged per LDS chunk
#define VSTR 40            // padded LDS row stride (floats): conflict-free B-frag reads
#define AV_LDS_BYTES (VCH * VSTR * 4)  // 40 KB dynamic LDS

#if defined(__has_builtin)
#if __has_builtin(__builtin_amdgcn_global_load_async_to_lds_b128) && \
    __has_builtin(__builtin_amdgcn_s_wait_asynccnt)
#define USE_ASYNC_LDS 1
#endif
#endif

__device__ __forceinline__ v8f wmma_f32(v2f a, v2f b, v8f c) {
  // 8 args: (neg_a, A, neg_b, B, c_mod, C, reuse_a, reuse_b)
  return __builtin_amdgcn_wmma_f32_16x16x4_f32(false, a, false, b, (short)0, c,
                                               false, false);
}

__device__ __forceinline__ float wave_max32(float v) {
  #pragma unroll
  for (int m = 16; m > 0; m >>= 1) v = fmaxf(v, __shfl_xor(v, m, 32));
  return v;
}
__device__ __forceinline__ float wave_sum32(float v) {
  #pragma unroll
  for (int m = 16; m > 0; m >>= 1) v += __shfl_xor(v, m, 32);
  return v;
}

// ---------------------------------------------------------------------------
// 1) Projections: out = X @ W^T for q/k/v, scattered to [B,H,N,Dh].
//    grid (ROWS/16, NC/64, 3), block 128 (4 waves, one 16x16 col tile each).
// ---------------------------------------------------------------------------
__global__ void __launch_bounds__(128) proj_kernel(
    const float* __restrict__ Xq, const float* __restrict__ Xk,
    const float* __restrict__ Xv, const float* __restrict__ Wq,
    const float* __restrict__ Wk, const float* __restrict__ Wv,
    float* __restrict__ Oq, float* __restrict__ Ok, float* __restrict__ Ov) {
  const int which = blockIdx.z;
  const float* X = (which == 0) ? Xq : ((which == 1) ? Xk : Xv);
  const float* W = (which == 0) ? Wq : ((which == 1) ? Wk : Wv);
  float* O = (which == 0) ? Oq : ((which == 1) ? Ok : Ov);

  const int wid = threadIdx.x >> 5;
  const int lane = threadIdx.x & 31;
  const int lr = lane & 15;
  const int ks = (lane >> 4) << 1;  // 0 for lanes 0-15, 2 for lanes 16-31
  const int r0 = blockIdx.x * 16;
  const int c0 = blockIdx.y * 64 + wid * 16;

  const float* xrow = X + (size_t)(r0 + lr) * NC;
  const float* wrow = W + (size_t)(c0 + lr) * NC;  // B[k][c'] = W[c'][k]

  v8f acc = {0.f, 0.f, 0.f, 0.f, 0.f, 0.f, 0.f, 0.f};
  #pragma unroll 8
  for (int kb = 0; kb < NC; kb += 4) {
    v2f a = *(const v2f*)(xrow + kb + ks);
    v2f b = *(const v2f*)(wrow + kb + ks);
    acc = wmma_f32(a, b, acc);
  }
  #pragma unroll
  for (int r = 0; r < 8; ++r) {
    const int rowm = (lane < 16) ? r : (r + 8);
    const int g = r0 + rowm;       // flat (n,b) row: g = n*NB + b
    const int n = g >> 2, b = g & 3;
    const int cfull = c0 + lr;
    const int h = cfull >> 5, d = cfull & 31;
    O[(((size_t)(b * NH + h)) * NT + n) * DH + d] = acc[r];
  }
}

// ---------------------------------------------------------------------------
// 2) Cosine normalization of q,k in place: one wave per (b,h,n) row of 32.
// ---------------------------------------------------------------------------
__global__ void __launch_bounds__(256) norm_kernel(float* __restrict__ Q,
                                                   float* __restrict__ K) {
  const int gw = (int)((blockIdx.x * blockDim.x + threadIdx.x) >> 5);
  const int lane = threadIdx.x & 31;
  float* base = (gw < NB * NH * NT) ? Q : K;
  const int r = gw & (NB * NH * NT - 1);
  float v = base[(size_t)r * DH + lane];
  float s = wave_sum32(v * v);
  base[(size_t)r * DH + lane] = v * rsqrtf(s);
}

// ---------------------------------------------------------------------------
// 3) Position log-bias: logpos[h,n,m] = log(relu(emb(n,m).pw[h] + pb[h])+1e-6)
//    grid (NT/256, NT), block 256. Shared over batch, computed once.
// ---------------------------------------------------------------------------
__global__ void __launch_bounds__(256) pos_kernel(
    const float* __restrict__ qb, const float* __restrict__ kbx,
    const float* __restrict__ pw, const float* __restrict__ pbias,
    float* __restrict__ logpos) {
  __shared__ float spw[NH * 64];
  __shared__ float spb[NH];
  const int tid = threadIdx.x;
  for (int i = tid; i < NH * 64; i += 256) spw[i] = pw[i];
  if (tid < NH) spb[tid] = pbias[tid];
  __syncthreads();

  const int n = blockIdx.y;
  const int m = blockIdx.x * 256 + tid;

  const float qx1 = qb[n * 4 + 0], qy1 = qb[n * 4 + 1];
  const float qx2 = qb[n * 4 + 2], qy2 = qb[n * 4 + 3];
  const float kx1 = kbx[m * 4 + 0], ky1 = kbx[m * 4 + 1];
  const float kx2 = kbx[m * 4 + 2], ky2 = kbx[m * 4 + 3];

  const float w = qx2 - qx1 + 1.f, h = qy2 - qy1 + 1.f;
  const float cx = 0.5f * (qx1 + qx2), cy = 0.5f * (qy1 + qy2);
  const float wr = kx2 - kx1 + 1.f, hr = ky2 - ky1 + 1.f;
  const float cxr = 0.5f * (kx1 + kx2), cyr = 0.5f * (ky1 + ky2);

  float comp[4];
  comp[0] = logf(fabsf((cx - cxr) / w) + 1e-3f) * 100.f;
  comp[1] = logf(fabsf((cy - cyr) / h) + 1e-3f) * 100.f;
  comp[2] = logf(w / wr) * 100.f;
  comp[3] = logf(h / hr) * 100.f;

  float invd[8];
  {
    float d = 1.f;
    const float rr = 2.3713737056616552f;  // 1000^(1/8)
    #pragma unroll
    for (int j = 0; j < 8; ++j) { invd[j] = 1.f / d; d *= rr; }
  }

  float acc[NH];
  #pragma unroll
  for (int hh = 0; hh < NH; ++hh) acc[hh] = spb[hh];

  #pragma unroll
  for (int c = 0; c < 4; ++c) {
    #pragma unroll
    for (int j = 0; j < 8; ++j) {
      float s, co;
      __sincosf(comp[c] * invd[j], &s, &co);
      #pragma unroll
      for (int hh = 0; hh < NH; ++hh)
        acc[hh] += s * spw[hh * 64 + c * 16 + j] +
                   co * spw[hh * 64 + c * 16 + 8 + j];
    }
  }
  #pragma unroll
  for (int hh = 0; hh < NH; ++hh) {
    const float p = fmaxf(acc[hh], 0.f);
    logpos[(((size_t)hh) * NT + n) * NT + m] = logf(p + 1e-6f);
  }
}

// ---------------------------------------------------------------------------
// 4) Raw scores S = q k^T, one 16x16 tile per wave (8 WMMAs over K=32).
//    grid (NT/64, NT/16, NB*NH), block 128.
// ---------------------------------------------------------------------------
__global__ void __launch_bounds__(128) score_kernel(
    const float* __restrict__ Q, const float* __restrict__ K,
    float* __restrict__ attn) {
  const int bh = blockIdx.z;
  const int wid = threadIdx.x >> 5;
  const int lane = threadIdx.x & 31;
  const int lr = lane & 15;
  const int ks = (lane >> 4) << 1;
  const int n0 = blockIdx.y * 16;
  const int m0 = (blockIdx.x * 4 + wid) * 16;

  const float* qr = Q + (size_t)bh * NT * DH + (size_t)(n0 + lr) * DH;
  const float* kr = K + (size_t)bh * NT * DH + (size_t)(m0 + lr) * DH;

  v8f acc = {0.f, 0.f, 0.f, 0.f, 0.f, 0.f, 0.f, 0.f};
  #pragma unroll
  for (int kk = 0; kk < DH; kk += 4) {
    v2f a = *(const v2f*)(qr + kk + ks);
    v2f b = *(const v2f*)(kr + kk + ks);
    acc = wmma_f32(a, b, acc);
  }
  float* arow = attn + (size_t)bh * NT * NT;
  #pragma unroll
  for (int r = 0; r < 8; ++r) {
    const int rowm = (lane < 16) ? r : (r + 8);
    arow[(size_t)(n0 + rowm) * NT + m0 + lr] = acc[r];
  }
}

// ---------------------------------------------------------------------------
// 5) Row softmax + log-pos bias, in place in attn (L2-resident).
//    One wave per row; 32 elements per lane kept in registers.
// ---------------------------------------------------------------------------
__global__ void __launch_bounds__(256) softmax_kernel(
    float* __restrict__ attn, const float* __restrict__ logpos) {
  const int gw = (int)((blockIdx.x * blockDim.x + threadIdx.x) >> 5);  // row id
  const int lane = threadIdx.x & 31;
  const int bh = gw >> 10;
  const int n = gw & (NT - 1);
  const int h = bh & (NH - 1);

  float* row = attn + (size_t)gw * NT;
  const float* lp = logpos + ((size_t)h * NT + n) * NT;
  __builtin_prefetch(lp + lane * 4, 0, 0);  // warm L2/WGP$ for bias row

  float s[32];
  float mx = -3.4e38f;
  #pragma unroll
  for (int j = 0; j < 32; ++j) {
    s[j] = row[lane + j * 32];
    mx = fmaxf(mx, s[j]);
  }
  mx = wave_max32(mx);
  float sum = 0.f;
  #pragma unroll
  for (int j = 0; j < 32; ++j) {
    s[j] = __expf(s[j] - mx);
    sum += s[j];
  }
  sum = wave_sum32(sum);
  const float inv = 1.f / sum;
  #pragma unroll
  for (int j = 0; j < 32; ++j)
    row[lane + j * 32] = s[j] * inv + lp[lane + j * 32];
}

// ---------------------------------------------------------------------------
// 6) X = attn @ V, one wave -> 16 rows x 32 cols (two C tiles share A frags).
//    All 4 waves in a WG share one bh, so V is staged into LDS once per WG
//    (async global->LDS when available), in 256-row chunks with a 40-float
//    padded stride so every 32-lane ds_load_b32 is bank-conflict free.
//    grid (NT/64, NB*NH), block 128, dynamic LDS = 40 KB.
// ---------------------------------------------------------------------------
__global__ void __launch_bounds__(128) av_kernel(
    const float* __restrict__ attn, const float* __restrict__ V,
    float* __restrict__ xout) {
  extern __shared__ float sV[];  // [VCH][VSTR]
  const int bh = blockIdx.y;
  const int wid = threadIdx.x >> 5;
  const int lane = threadIdx.x & 31;
  const int lr = lane & 15;
  const int ks = (lane >> 4) << 1;
  const int n0 = (blockIdx.x * 4 + wid) * 16;

  const float* prow = attn + (size_t)bh * NT * NT + (size_t)(n0 + lr) * NT;
  const float* v = V + (size_t)bh * NT * DH;

  v8f acc0 = {0.f, 0.f, 0.f, 0.f, 0.f, 0.f, 0.f, 0.f};
  v8f acc1 = {0.f, 0.f, 0.f, 0.f, 0.f, 0.f, 0.f, 0.f};

  for (int c0 = 0; c0 < NT; c0 += VCH) {
    if (c0) __syncthreads();  // previous chunk fully consumed before overwrite
    // Stage VCH x 32 floats (32 KB payload): 2048 float4's over 128 threads.
    for (int i = threadIdx.x; i < (VCH * DH) / 4; i += 128) {
      const int m = i >> 3;            // V row within chunk
      const int dc = (i & 7) * 4;      // 4-float column chunk
      const float* g = v + (size_t)(c0 + m) * DH + dc;
      float* l = sV + m * VSTR + dc;
#if defined(USE_ASYNC_LDS)
      __builtin_amdgcn_global_load_async_to_lds_b128(
          (GLOBAL_AS v4i*)g, (LDS_AS v4i*)l, 0, 0);
#else
      *(v4f*)l = *(const v4f*)g;
#endif
    }
#if defined(USE_ASYNC_LDS)
    __builtin_amdgcn_s_wait_asynccnt(0);
#endif
    __syncthreads();

    #pragma unroll 4
    for (int kk = 0; kk < VCH; kk += 4) {
      v2f a = *(const v2f*)(prow + c0 + kk + ks);
      const float* vb = sV + (kk + ks) * VSTR;
      v2f b0, b1;
      b0.x = vb[lr];        b0.y = vb[VSTR + lr];       // cols d = 0..15
      b1.x = vb[16 + lr];   b1.y = vb[VSTR + 16 + lr];  // cols d = 16..31
      acc0 = wmma_f32(a, b0, acc0);
      acc1 = wmma_f32(a, b1, acc1);
    }
  }

  const int b = bh >> 3, h = bh & 7;
  #pragma unroll
  for (int r = 0; r < 8; ++r) {
    const int rowm = (lane < 16) ? r : (r + 8);
    const int n = n0 + rowm;
    float* xr = xout + ((size_t)n * NB + b) * NC + h * DH;
    xr[lr] = acc0[r];
    xr[16 + lr] = acc1[r];
  }
}

// ---------------------------------------------------------------------------
extern "C" void kernel_launch(void* const* d_in, const int* in_sizes, int n_in,
                              void* d_out, int out_size, void* d_ws,
                              size_t ws_size, hipStream_t stream) {
  (void)in_sizes; (void)n_in; (void)out_size; (void)ws_size;
  const float* query = (const float*)d_in[0];
  const float* key_  = (const float*)d_in[1];
  const float* value = (const float*)d_in[2];
  const float* qbox  = (const float*)d_in[3];
  const float* kbox  = (const float*)d_in[4];
  const float* Wq    = (const float*)d_in[5];
  const float* Wk    = (const float*)d_in[6];
  const float* Wv    = (const float*)d_in[7];
  const float* pw    = (const float*)d_in[8];
  const float* pb    = (const float*)d_in[9];

  // Workspace: qn | kn | vv (4 MB each, [B,H,N,Dh]) | logpos (32 MB, [H,N,N])
  float* ws = (float*)d_ws;
  const size_t proj_elems = (size_t)NB * NH * NT * DH;
  float* qn = ws;
  float* kn = qn + proj_elems;
  float* vv = kn + proj_elems;
  float* lp = vv + proj_elems;

  float* xout = (float*)d_out;          // [N,B,C]
  float* attn = xout + XSIZE;           // [B,H,N,N]

  proj_kernel<<<dim3(ROWS / 16, NC / 64, 3), 128, 0, stream>>>(
      query, key_, value, Wq, Wk, Wv, qn, kn, vv);
  norm_kernel<<<dim3((2 * NB * NH * NT) / 8), 256, 0, stream>>>(qn, kn);
  pos_kernel<<<dim3(NT / 256, NT), 256, 0, stream>>>(qbox, kbox, pw, pb, lp);
  score_kernel<<<dim3(NT / 64, NT / 16, NB * NH), 128, 0, stream>>>(qn, kn,
                                                                    attn);
  softmax_kernel<<<dim3((NB * NH * NT) / 8), 256, 0, stream>>>(attn, lp);
  av_kernel<<<dim3(NT / 64, NB * NH), 128, AV_LDS_BYTES, stream>>>(attn, vv,
                                                                   xout);
}